// AttentionConstrainedLoss_29532195127820
// MI455X (gfx1250) — compile-verified
//
#include <hip/hip_runtime.h>
#include <hip/hip_bf16.h>

typedef float v2f __attribute__((ext_vector_type(2)));
typedef float v8f __attribute__((ext_vector_type(8)));

#define HW   65536
#define NB   4
#define MBOX 100
#define DCH  128

// ---------------- Workspace layout (bytes) ----------------
// [0,            1048576)  vbuf      : float  [NB*HW]   per-pixel variance
// [1048576,      2097152)  claimcnt  : u32    [NB*HW]   # of claims (parity)
// [2097152,      3145728)  claimmax  : int    [NB*HW]   max claiming box index + 1 (0 = none)
// [3145728,      3147328)  sums      : float  [NB*MBOX]
// [3147328,      3148928)  cnts      : float  [NB*MBOX]

// ---------------- K0: zero claim/accumulator region ----------------
__global__ __launch_bounds__(256) void zero_kernel(unsigned* __restrict__ z, int n) {
    int stride = gridDim.x * blockDim.x;
    for (int i = blockIdx.x * blockDim.x + threadIdx.x; i < n; i += stride) z[i] = 0u;
}

// ---------------- K1: per-pixel variance via V_WMMA_F32_16X16X4_F32 ----------
// One wave handles a tile of 16 pixels x 128 channels. B matrix = all ones, so
// D[m][n] = row-sum of A for every n (exact f32). Two chains: sum(x), sum(x^2).
__global__ __launch_bounds__(256) void var_wmma_kernel(const float* __restrict__ atten,
                                                       float* __restrict__ vbuf) {
    const int lane = threadIdx.x & 31;
    const int wave = threadIdx.x >> 5;
    const int tile = blockIdx.x * 8 + wave;          // 16 pixels per tile
    const int row  = lane & 15;                      // A-matrix M row for this lane
    const int koff = (lane >> 4) << 1;               // K offset: 0 (lanes 0-15) / 2 (16-31)
    const float* rp = atten + (size_t)(tile * 16 + row) * DCH + koff;

    v8f accS = {0.f,0.f,0.f,0.f,0.f,0.f,0.f,0.f};
    v8f accQ = {0.f,0.f,0.f,0.f,0.f,0.f,0.f,0.f};
    v2f ones; ones.x = 1.0f; ones.y = 1.0f;

    #pragma unroll 8
    for (int k0 = 0; k0 < DCH; k0 += 4) {
        v2f a  = *(const v2f*)(rp + k0);
        v2f a2 = a * a;
        accS = __builtin_amdgcn_wmma_f32_16x16x4_f32(false, a,  false, ones, (short)0, accS, false, false);
        accQ = __builtin_amdgcn_wmma_f32_16x16x4_f32(false, a2, false, ones, (short)0, accQ, false, false);
    }
    // C/D layout: VGPR r -> M=r (lanes 0-15, N=lane) / M=8+r (lanes 16-31).
    // Column 0 lives in lanes 0 and 16.
    if ((lane & 15) == 0) {
        const int half = lane >> 4;
        #pragma unroll
        for (int r = 0; r < 8; ++r) {
            float s = accS[r];
            float q = accQ[r];
            float v = (q - s * s * (1.0f / 128.0f)) * (1.0f / 127.0f);  // ddof=1
            vbuf[tile * 16 + half * 8 + r] = v;
        }
    }
}

// ---------------- K2: per-box rasterize + claim (one wave32 per box) ---------
__global__ __launch_bounds__(256) void box_claim_kernel(const float* __restrict__ boxes,
                                                        unsigned* __restrict__ claimcnt,
                                                        int* __restrict__ claimmax) {
    const int lane = threadIdx.x & 31;
    const int t    = blockIdx.x * 8 + (threadIdx.x >> 5);   // global box id 0..399
    const int b    = t / MBOX;
    const int i    = t - b * MBOX;

    const float* bx = boxes + (size_t)t * 7;
    const float cx = bx[0], cy = bx[1], l = bx[3], w = bx[4], yaw = bx[6];

    // effective dims: ratio = clip((DIMS/W)/lw, 1, 6); eff = lw * ratio
    const float rl = fminf(fmaxf(0.4f / l, 1.0f), 6.0f);
    const float rw = fminf(fmaxf(0.4f / w, 1.0f), 6.0f);
    const float el = l * rl, ew = w * rw;
    const float c = cosf(yaw), s = sinf(yaw);

    const float cnx[4] = {-0.5f, -0.5f, 0.5f, 0.5f};
    const float cny[4] = {-0.5f,  0.5f, 0.5f, -0.5f};
    float px[4], py[4];
    #pragma unroll
    for (int k = 0; k < 4; ++k) {
        float ux = el * cnx[k], uy = ew * cny[k];
        px[k] =  ux * c + uy * s + cx;
        py[k] = -ux * s + uy * c + cy;
    }
    float ex[4], ey[4];
    #pragma unroll
    for (int k = 0; k < 4; ++k) {
        int kn = (k + 1) & 3;
        ex[k] = px[kn] - px[k];
        ey[k] = py[kn] - py[k];
    }

    // nearest grid cell (replicates argmin first-index tie-break on regular grid)
    const float tx = (cx + 51.2f) * 2.5f;
    const float ty = (cy + 51.2f) * 2.5f;
    int nix = (int)ceilf(tx) - 1; nix = min(max(nix, 0), 255);
    int niy = (int)ceilf(ty) - 1; niy = min(max(niy, 0), 255);
    const int np = niy * 256 + nix;

    // bounding box in grid indices (widened by 1; exact cross test decides)
    float minx = fminf(fminf(px[0], px[1]), fminf(px[2], px[3]));
    float maxx = fmaxf(fmaxf(px[0], px[1]), fmaxf(px[2], px[3]));
    float miny = fminf(fminf(py[0], py[1]), fminf(py[2], py[3]));
    float maxy = fmaxf(fmaxf(py[0], py[1]), fmaxf(py[2], py[3]));
    int i0 = min(max((int)floorf((minx + 51.2f) * 2.5f - 0.5f) - 1, 0), 255);
    int i1 = min(max((int)ceilf ((maxx + 51.2f) * 2.5f - 0.5f) + 1, 0), 255);
    int j0 = min(max((int)floorf((miny + 51.2f) * 2.5f - 0.5f) - 1, 0), 255);
    int j1 = min(max((int)ceilf ((maxy + 51.2f) * 2.5f - 0.5f) + 1, 0), 255);
    const int nx = i1 - i0 + 1;
    const int ny = j1 - j0 + 1;
    const int tot = nx * ny;

    unsigned* cc = claimcnt + (size_t)b * HW;
    int*      cm = claimmax + (size_t)b * HW;

    for (int u = lane; u < tot; u += 32) {
        const int ix = i0 + (u % nx);
        const int iy = j0 + (u / nx);
        const float gx = ((float)ix + 0.5f) * 0.4f - 51.2f;
        const float gy = ((float)iy + 0.5f) * 0.4f - 51.2f;
        float cr0 = ex[0] * (gy - py[0]) - ey[0] * (gx - px[0]);
        float cr1 = ex[1] * (gy - py[1]) - ey[1] * (gx - px[1]);
        float cr2 = ex[2] * (gy - py[2]) - ey[2] * (gx - px[2]);
        float cr3 = ex[3] * (gy - py[3]) - ey[3] * (gx - px[3]);
        bool ge = (cr0 >= 0.f) && (cr1 >= 0.f) && (cr2 >= 0.f) && (cr3 >= 0.f);
        bool le = (cr0 <= 0.f) && (cr1 <= 0.f) && (cr2 <= 0.f) && (cr3 <= 0.f);
        const int q = iy * 256 + ix;
        if (ge || le || (q == np)) {
            atomicAdd(&cc[q], 1u);
            atomicMax(&cm[q], i + 1);
        }
    }
    // nearest cell outside the bbox window still gets its claim
    if (lane == 0 && (nix < i0 || nix > i1 || niy < j0 || niy > j1)) {
        atomicAdd(&cc[np], 1u);
        atomicMax(&cm[np], i + 1);
    }
}

// ---------------- K3: resolve toggle-scan + segment accumulate ---------------
// flag = (claim parity odd) ? max claiming index : -1  (matches the scan)
__global__ __launch_bounds__(256) void resolve_kernel(const float* __restrict__ vbuf,
                                                      const unsigned* __restrict__ claimcnt,
                                                      const int* __restrict__ claimmax,
                                                      float* __restrict__ sums,
                                                      float* __restrict__ cnts) {
    const int q = blockIdx.x * 256 + threadIdx.x;   // 0 .. NB*HW-1
    const unsigned cnt = claimcnt[q];
    if (cnt & 1u) {
        const int flag = claimmax[q] - 1;           // guaranteed >= 0 when cnt>0
        const int b = q >> 16;                      // HW = 65536
        const float v = vbuf[q];
        atomicAdd(&sums[b * MBOX + flag], v);
        atomicAdd(&cnts[b * MBOX + flag], 1.0f);
    }
}

// ---------------- K4: final reduction to scalar ------------------------------
__global__ __launch_bounds__(512) void finalize_kernel(const float* __restrict__ sums,
                                                       const float* __restrict__ cnts,
                                                       float* __restrict__ out) {
    __shared__ float ls[512];
    __shared__ float ln[512];
    const int t = threadIdx.x;
    float lsum = 0.f, lnum = 0.f;
    if (t < NB * MBOX) {
        const float c = cnts[t];
        if (c > 0.f) {
            lsum = sums[t] / fmaxf(c, 1.0f);
            lnum = 1.0f;
        }
    }
    ls[t] = lsum; ln[t] = lnum;
    __syncthreads();
    for (int s = 256; s > 0; s >>= 1) {
        if (t < s) { ls[t] += ls[t + s]; ln[t] += ln[t + s]; }
        __syncthreads();
    }
    if (t == 0) out[0] = -ls[0] / fmaxf(ln[0], 1.0f);
}

extern "C" void kernel_launch(void* const* d_in, const int* in_sizes, int n_in,
                              void* d_out, int out_size, void* d_ws, size_t ws_size,
                              hipStream_t stream) {
    (void)in_sizes; (void)n_in; (void)out_size; (void)ws_size;
    const float* atten = (const float*)d_in[0];   // (4, 65536, 128) f32
    const float* boxes = (const float*)d_in[1];   // (4, 100, 7)    f32
    // d_in[2] = gt_labels, unused by the reference loss

    char* ws = (char*)d_ws;
    float*    vbuf     = (float*)   (ws + 0);
    unsigned* claimcnt = (unsigned*)(ws + 1048576);
    int*      claimmax = (int*)     (ws + 2097152);
    float*    sums     = (float*)   (ws + 3145728);
    float*    cnts     = (float*)   (ws + 3147328);

    // zero region [claimcnt .. cnts] = 2*262144 + 2*400 u32 words, contiguous
    zero_kernel<<<512, 256, 0, stream>>>(claimcnt, 2 * NB * HW + 2 * NB * MBOX);

    // per-pixel variance (memory-bound long pole, ~134 MB -> ~6 us)
    var_wmma_kernel<<<(NB * HW / 16) / 8, 256, 0, stream>>>(atten, vbuf);

    // per-box rasterization: 400 boxes, one wave each -> 50 blocks of 8 waves
    box_claim_kernel<<<50, 256, 0, stream>>>(boxes, claimcnt, claimmax);

    // resolve flags + segment sums
    resolve_kernel<<<NB * HW / 256, 256, 0, stream>>>(vbuf, claimcnt, claimmax, sums, cnts);

    // scalar loss
    finalize_kernel<<<1, 512, 0, stream>>>(sums, cnts, (float*)d_out);
}